// InformationGainObjectDetection_15169824489654
// MI455X (gfx1250) — compile-verified
//
#include <hip/hip_runtime.h>
#include <cstdint>
#include <cstddef>

// ---------------- problem constants (match reference setup_inputs) ----------
#define K_W   4096
#define K_H   2048
#define K_WS  2048      // W / SUBSAMPLE
#define K_HS  1024      // H / SUBSAMPLE
#define K_HG  16
#define K_WG  32
#define NRMAX 1536      // 2*Nc + Np

// ---------------- CDNA5 feature detection -----------------------------------
typedef float v2f __attribute__((ext_vector_type(2)));
typedef float v8f __attribute__((ext_vector_type(8)));
typedef int   v4i __attribute__((ext_vector_type(4)));

#define AS1 __attribute__((address_space(1)))
#define AS3 __attribute__((address_space(3)))

#if defined(__gfx1250__) && __has_builtin(__builtin_amdgcn_global_load_async_to_lds_b128)
#define HAVE_ASYNC 1
#else
#define HAVE_ASYNC 0
#endif

#if defined(__gfx1250__) && __has_builtin(__builtin_amdgcn_wmma_f32_16x16x4_f32)
#define HAVE_WMMA 1
#else
#define HAVE_WMMA 0
#endif

#if HAVE_ASYNC
__device__ __forceinline__ void wait_async0() {
#if __has_builtin(__builtin_amdgcn_s_wait_asynccnt)
  __builtin_amdgcn_s_wait_asynccnt(0);
#else
  asm volatile("s_wait_asynccnt 0" ::: "memory");
#endif
}
#endif

// ---------------- helpers ---------------------------------------------------
__device__ __forceinline__ int4 grect_i(int x1, int y1, int x2, int y2) {
  // block_size = 128 -> >>7 (coords are non-negative)
  return make_int4(x1 >> 7, y1 >> 7, (x2 - 1) >> 7, (y2 - 1) >> 7);
}
__device__ __forceinline__ bool rovl(const int4 a, const int4 b) {
  return (a.x <= b.z) && (b.x <= a.z) && (a.y <= b.w) && (b.y <= a.w);
}

// ============================================================================
// Kernel 1: per-box prep. One block, 512 threads (thread i = curr box i and
// prev box i). Produces the two flattened rect tables in workspace:
//   rm[0..2Nc+Np) / pv : painting rects (cell coords, half-open) + premultiplied
//                     value v*act  (act folded: inactive -> value 0)
//   rw[0..Nc+Np) / wv : grid_ig rects (grid coords, inclusive); inactive prev
//                     rects encoded as degenerate (cover nothing)
// ============================================================================
__global__ __launch_bounds__(512)
void ig_prep(const float* __restrict__ boxc, const float* __restrict__ boxp,
             int Nc, int Np,
             int4* __restrict__ rm, float* __restrict__ pv,
             int4* __restrict__ rw, float* __restrict__ wv)
{
  __shared__ float4 s_cpf[512];  // float(int(prev/2)) coords
  __shared__ int4   s_gp[512];   // prev grid rects
  __shared__ int4   s_gc[512];   // curr grid rects
  __shared__ float  s_sp[512];   // prev scores
  __shared__ int    s_mp[512];   // matched_prev flags

  const int i = threadIdx.x;

  int4   cc_i = make_int4(0, 0, 0, 0), gc_i = make_int4(0, 0, 0, 0);
  float4 a = make_float4(0.f, 0.f, 0.f, 0.f);
  float  areaA = 0.f, csc = 0.f;
  bool   big = false;

  if (i < Nc) {
    float x1 = boxc[i * 5 + 0], y1 = boxc[i * 5 + 1];
    float x2 = boxc[i * 5 + 2], y2 = boxc[i * 5 + 3];
    csc = boxc[i * 5 + 4];
    cc_i  = make_int4((int)(x1 * 0.5f), (int)(y1 * 0.5f),
                      (int)(x2 * 0.5f), (int)(y2 * 0.5f));
    a     = make_float4((float)cc_i.x, (float)cc_i.y, (float)cc_i.z, (float)cc_i.w);
    areaA = (a.z - a.x) * (a.w - a.y);
    gc_i  = grect_i((int)x1, (int)y1, (int)x2, (int)y2);
    big   = ((int)y2 - (int)y1) >= 100;
    s_gc[i] = gc_i;
  }
  if (i < Np) {
    float x1 = boxp[i * 5 + 0], y1 = boxp[i * 5 + 1];
    float x2 = boxp[i * 5 + 2], y2 = boxp[i * 5 + 3];
    int4 cp_i = make_int4((int)(x1 * 0.5f), (int)(y1 * 0.5f),
                          (int)(x2 * 0.5f), (int)(y2 * 0.5f));
    s_cpf[i] = make_float4((float)cp_i.x, (float)cp_i.y, (float)cp_i.z, (float)cp_i.w);
    s_gp[i]  = grect_i((int)x1, (int)y1, (int)x2, (int)y2);
    s_sp[i]  = boxp[i * 5 + 4];
    s_mp[i]  = 0;
  }
  __syncthreads();

  // ---- best IoU match (argmax = first max, like jnp.argmax) ----
  float best = -1.f; int bj = 0;
  if (i < Nc) {
    for (int j = 0; j < Np; ++j) {
      float4 b  = s_cpf[j];
      float ix1 = fmaxf(a.x, b.x), iy1 = fmaxf(a.y, b.y);
      float ix2 = fminf(a.z, b.z), iy2 = fminf(a.w, b.w);
      float inter = fmaxf(ix2 - ix1, 0.f) * fmaxf(iy2 - iy1, 0.f);
      float areaB = (b.z - b.x) * (b.w - b.y);
      float iou   = inter / (areaA + areaB - inter);
      if (iou > best) { best = iou; bj = j; }
    }
  }
  const bool  matched = (i < Nc) && (best > 0.f);
  const float igv     = 1.f - best;

  if (matched) atomicOr(&s_mp[bj], 1);

  // ---- isolation test over all 1024 grid rects ----
  bool covered = false;
  int4 gpj = make_int4(0, 0, 0, 0);
  if (i < Nc) {
    gpj = s_gp[bj];
    const int tot = Nc + Np;
    for (int k = 0; k < tot; ++k) {
      int4 q  = (k < Nc) ? s_gc[k] : s_gp[k - Nc];
      bool ov = rovl(gc_i, q) || (matched && rovl(gpj, q));
      bool ex = (k == i) || (matched && (k == Nc + bj));
      covered |= (ov && !ex);
    }
  }
  __syncthreads();   // s_mp final

  if (i < Nc) {
    float val_c = (!covered && big && (csc >= 0.7f)) ? 2.f : 1.f;
    // paint segment 0: cc, ig*sc (act=1)
    rm[i] = cc_i;
    pv[i] = igv * csc;
    // paint segment 1: cp[best_j], ig*sp[best_j]*matched
    float4 cb = s_cpf[bj];
    rm[Nc + i] = make_int4((int)cb.x, (int)cb.y, (int)cb.z, (int)cb.w);
    pv[Nc + i] = matched ? (igv * s_sp[bj]) : 0.f;
    // grid_ig segment 0: wrect (union if matched), val_c
    int4 u = matched ? make_int4(min(gc_i.x, gpj.x), min(gc_i.y, gpj.y),
                                 max(gc_i.z, gpj.z), max(gc_i.w, gpj.w))
                     : gc_i;
    rw[i] = u;
    wv[i] = val_c;
  }
  if (i < Np) {
    // paint segment 2: cp, sp * (~matched_prev)
    float4 cf = s_cpf[i];
    rm[2 * Nc + i] = make_int4((int)cf.x, (int)cf.y, (int)cf.z, (int)cf.w);
    pv[2 * Nc + i] = s_mp[i] ? 0.f : s_sp[i];
    // grid_ig segment 1: gp, val 1, active iff ~matched_prev (degenerate if not)
    rw[Nc + i] = s_mp[i] ? make_int4(1 << 28, 1 << 28, -(1 << 28), -(1 << 28))
                         : s_gp[i];
    wv[Nc + i] = 1.f;
  }
}

// ============================================================================
// Kernel 2: grid_ig (16 x 32). "last active covering rect wins" (inclusive).
// ============================================================================
__global__ __launch_bounds__(512)
void ig_grid(const int4* __restrict__ rw, const float* __restrict__ wv, int nW,
             float* __restrict__ outg)
{
  const int t = threadIdx.x;
  if (t >= K_HG * K_WG) return;
  const int y = t >> 5, x = t & 31;
  float val = 0.f;
  for (int r = 0; r < nW; ++r) {
    int4 q = rw[r];
    if (x >= q.x && x <= q.z && y >= q.y && y <= q.w) val = wv[r];
  }
  outg[t] = val;
}

// ============================================================================
// Kernel 3: max-paint m (1024 x 2048) and write the 2x-upsampled mask.
// Block = 256 threads = 8 waves covering a 32(x) x 64(y) m-cell region;
// each wave owns one 16x16 m-tile, painted directly in WMMA B-layout
// (lane L=lane>>4, c=lane&15 holds rows 4k+2L and 4k+2L+1 of column c),
// then out = E * m (E = 0/1 row-replication matrix as the A operand) via
// 8 chained v_wmma_f32_16x16x4_f32; column doubling happens at store time.
// Rect table staged to LDS with CDNA5 async load-to-LDS, then compacted.
// ============================================================================
__global__ __launch_bounds__(256)
void ig_paint(const int4* __restrict__ rm, const float* __restrict__ pv, int nR,
              float* __restrict__ mask)
{
  __shared__ int4  srect[NRMAX];
  __shared__ float sval[NRMAX];
  __shared__ int4  frect[NRMAX];
  __shared__ float fval[NRMAX];
  __shared__ int   fcount;

  const int tid = threadIdx.x;
  if (tid == 0) fcount = 0;

  // ---- stage full rect table into LDS (CDNA5 async path, ASYNCcnt) ----
#if HAVE_ASYNC
  for (int k = tid; k < nR; k += 256)
    __builtin_amdgcn_global_load_async_to_lds_b128(
        (AS1 v4i*)(AS1 void*)(rm + k),
        (AS3 v4i*)(AS3 void*)(srect + k), 0, 0);
  for (int k = tid; 4 * k < nR; k += 256)   // nR is a multiple of 4
    __builtin_amdgcn_global_load_async_to_lds_b128(
        (AS1 v4i*)(AS1 void*)(pv + 4 * k),
        (AS3 v4i*)(AS3 void*)(sval + 4 * k), 0, 0);
  wait_async0();
#else
  for (int k = tid; k < nR; k += 256) { srect[k] = rm[k]; sval[k] = pv[k]; }
#endif
  __syncthreads();

  // ---- compact rects overlapping this block's region into LDS list ----
  const int Rx0 = blockIdx.x * 32;
  const int Ry0 = blockIdx.y * 64;
  for (int k = tid; k < nR; k += 256) {
    int4  q = srect[k];
    float v = sval[k];
    if (v > 0.f && q.x < Rx0 + 32 && q.z > Rx0 && q.y < Ry0 + 64 && q.w > Ry0) {
      int idx = atomicAdd(&fcount, 1);
      frect[idx] = q;
      fval[idx]  = v;
    }
  }
  __syncthreads();
  const int n = fcount;

  // ---- per-wave 16x16 tile, painted in WMMA B-layout ----
  const int lane = tid & 31;
  const int w    = tid >> 5;
  const int mx = Rx0 + (w & 1) * 16;   // m-tile origin x
  const int my = Ry0 + (w >> 1) * 16;  // m-tile origin y
  const int L = lane >> 4;
  const int c = lane & 15;
  const int x = mx + c;                // fixed m-column per lane

  float bv[8];
  int   yr[8];
#pragma unroll
  for (int v = 0; v < 8; ++v) {
    bv[v] = 0.f;
    yr[v] = my + 4 * (v >> 1) + 2 * L + (v & 1);  // B-layout row of slot v
  }
  for (int r = 0; r < n; ++r) {
    int4 q = frect[r];
    if (x >= q.x && x < q.z) {
      float p = fval[r];
#pragma unroll
      for (int v = 0; v < 8; ++v)
        if (yr[v] >= q.y && yr[v] < q.w) bv[v] = fmaxf(bv[v], p);
    }
  }

#if HAVE_WMMA
  // out_top(16x16) = E0 * m, out_bot = E1 * m ; E{0,1}[i][j] = ((i{+16})>>1==j)
  // A chunk k: lane holds M=c, K = 4k+2L and 4k+2L+1 (pure lane-id constants).
  v8f ct = {};  v8f cb = {};
  const int tgtT = c >> 1;
  const int tgtB = (c >> 1) + 8;
#pragma unroll
  for (int k = 0; k < 4; ++k) {
    const int Kg = 4 * k + 2 * L;
    v2f bb; bb.x = bv[2 * k];            bb.y = bv[2 * k + 1];
    v2f at; at.x = (tgtT == Kg) ? 1.f : 0.f; at.y = (tgtT == Kg + 1) ? 1.f : 0.f;
    v2f ab; ab.x = (tgtB == Kg) ? 1.f : 0.f; ab.y = (tgtB == Kg + 1) ? 1.f : 0.f;
    ct = __builtin_amdgcn_wmma_f32_16x16x4_f32(false, at, false, bb, (short)0, ct, false, false);
    cb = __builtin_amdgcn_wmma_f32_16x16x4_f32(false, ab, false, bb, (short)0, cb, false, false);
  }
  // D layout: vgpr v -> out row v (lanes 0-15) / v+8 (lanes 16-31), col c.
  const int  ocol  = (mx + c) * 2;
  const long orow0 = (long)my * 2;
#pragma unroll
  for (int v = 0; v < 8; ++v) {
    const int i0 = v + 8 * L;
    float2 d0; d0.x = ct[v]; d0.y = ct[v];
    *(float2*)(mask + (orow0 + i0) * K_W + ocol) = d0;
    float2 d1; d1.x = cb[v]; d1.y = cb[v];
    *(float2*)(mask + (orow0 + 16 + i0) * K_W + ocol) = d1;
  }
#else
  const int ocol = (mx + c) * 2;
#pragma unroll
  for (int v = 0; v < 8; ++v) {
    const long gy = (long)yr[v] * 2;
    float2 d; d.x = bv[v]; d.y = bv[v];
    *(float2*)(mask + gy * K_W + ocol) = d;
    *(float2*)(mask + (gy + 1) * K_W + ocol) = d;
  }
#endif
}

// ============================================================================
// Host entry
// ============================================================================
extern "C" void kernel_launch(void* const* d_in, const int* in_sizes, int n_in,
                              void* d_out, int out_size, void* d_ws, size_t ws_size,
                              hipStream_t stream)
{
  (void)n_in; (void)out_size; (void)ws_size;

  const float* boxc = (const float*)d_in[0];
  const float* boxp = (const float*)d_in[1];
  int Nc = in_sizes[0] / 5;
  int Np = in_sizes[1] / 5;
  if (Nc > 512) Nc = 512;
  if (Np > 512) Np = 512;

  // workspace layout (all offsets 16B aligned), capacity for Nc=Np=512
  char*  ws = (char*)d_ws;
  int4*  rm = (int4*)(ws);                                   // 1536 * 16B
  float* pv = (float*)(ws + 1536 * 16);                      // 1536 * 4B
  int4*  rw = (int4*)(ws + 1536 * 16 + 1536 * 4);            // 1024 * 16B
  float* wv = (float*)(ws + 1536 * 16 + 1536 * 4 + 1024 * 16); // 1024 * 4B

  float* mask = (float*)d_out;                               // (1,1,2048,4096)
  float* grid = mask + (size_t)K_H * K_W;                    // (1,1,16,32)

  const int nR = 2 * Nc + Np;
  const int nW = Nc + Np;

  ig_prep <<<dim3(1), dim3(512), 0, stream>>>(boxc, boxp, Nc, Np, rm, pv, rw, wv);
  ig_grid <<<dim3(1), dim3(512), 0, stream>>>(rw, wv, nW, grid);
  ig_paint<<<dim3(K_WS / 32, K_HS / 64), dim3(256), 0, stream>>>(rm, pv, nR, mask);
}